// GNN_81088982548491
// MI455X (gfx1250) — compile-verified
//
#include <hip/hip_runtime.h>
#include <math.h>

typedef __attribute__((ext_vector_type(2))) float v2f;
typedef __attribute__((ext_vector_type(8))) float v8f;

#define N_NODES 50000
#define BN_EPS 1e-5f
#define MT 5   // 16-row tiles per wave; 50000 = 625 * (5*16) exactly

// ---------------- utility kernels ----------------

__global__ void fill_kernel(float* __restrict__ p, float v, int n) {
    int i = blockIdx.x * blockDim.x + threadIdx.x;
    if (i < n) p[i] = v;
}

__global__ void addvec_kernel(const float* __restrict__ a, const float* __restrict__ b,
                              float* __restrict__ o, int n) {
    int i = blockIdx.x * blockDim.x + threadIdx.x;
    if (i < n) o[i] = a[i] + b[i];
}

__global__ void degree_kernel(const int* __restrict__ dst, float* __restrict__ cnt, int E) {
    int e = blockIdx.x * blockDim.x + threadIdx.x;
    if (e < E) atomicAdd(&cnt[dst[e]], 1.0f);
}

// scatter-add neighbor features: agg[dst] += h[src], float4 per thread
__global__ void scatter_kernel(const float* __restrict__ h, const int* __restrict__ src,
                               const int* __restrict__ dst, float* __restrict__ agg,
                               int E, int din) {
    int q = din >> 2; // float4 chunks per row
    long total = (long)E * q;
    long tid = (long)blockIdx.x * blockDim.x + threadIdx.x;
    if (tid >= total) return;
    int e = (int)(tid / q);
    int f = (int)(tid % q) * 4;
    int s = src[e], d = dst[e];
    const float4 v = *(const float4*)(h + (long)s * din + f);
    float* o = agg + (long)d * din + f;
    atomicAdd(o + 0, v.x);
    atomicAdd(o + 1, v.y);
    atomicAdd(o + 2, v.z);
    atomicAdd(o + 3, v.w);
}

// ---------------- fused SAGE layer GEMM (WMMA f32 16x16x4) ----------------
// out[n, dout] = act( (agg[n,:]/cnt[n]) @ Wl + h[n,:] @ Wc + (bl+bs) ),  Wc = Wr+Ws
// One wave computes MT row-tiles x 16 cols: B fragment loaded once per K-step,
// reused across MT independent WMMA accumulator chains. BN column sum/sumsq
// partials are fused into the epilogue.
__global__ void __launch_bounds__(32)
sage_gemm_kernel(const float* __restrict__ agg, const float* __restrict__ h,
                 const float* __restrict__ cnt,
                 const float* __restrict__ Wl, const float* __restrict__ Wc,
                 const float* __restrict__ bl, const float* __restrict__ bs,
                 float* __restrict__ out, float* __restrict__ colsum,
                 float* __restrict__ colsq,
                 int din, int dout, int do_relu, int do_bn) {
    const int m0 = blockIdx.x * (16 * MT);
    const int n0 = blockIdx.y * 16;
    const int lane = threadIdx.x;
    const int half = lane >> 4;   // 0: lanes 0-15 (K=0,1), 1: lanes 16-31 (K=2,3)
    const int l = lane & 15;
    const int col = n0 + l;
    const bool colok = col < dout;
    const int colc = colok ? col : 0;

    float invc[MT];
    const float* arow[MT];
    const float* hrow[MT];
    #pragma unroll
    for (int t = 0; t < MT; ++t) {
        const int r = m0 + t * 16 + l;   // A fragment: lanes l and l+16 share row l
        invc[t] = 1.0f / fmaxf(cnt[r], 1.0f);
        arow[t] = agg + (long)r * din + 2 * half;
        hrow[t] = h + (long)r * din + 2 * half;
    }

    v8f c[MT];
    #pragma unroll
    for (int t = 0; t < MT; ++t) c[t] = (v8f){};

    // B fragment base: element (k + 2*half, colc), row-major [din x dout]
    const float* WlB = Wl + (long)(2 * half) * dout + colc;
    const float* WcB = Wc + (long)(2 * half) * dout + colc;

    // pass 1: (agg/cnt) @ Wl
    for (int k = 0; k < din; k += 4) {
        const float b0 = WlB[(long)k * dout];
        const float b1 = WlB[(long)k * dout + dout];
        v2f b;
        b.x = colok ? b0 : 0.0f;
        b.y = colok ? b1 : 0.0f;
        #pragma unroll
        for (int t = 0; t < MT; ++t) {
            float2 av = *(const float2*)(arow[t] + k);
            v2f a;
            a.x = av.x * invc[t];
            a.y = av.y * invc[t];
            c[t] = __builtin_amdgcn_wmma_f32_16x16x4_f32(false, a, false, b, (short)0,
                                                         c[t], false, false);
        }
    }

    // pass 2: h @ (Wr + Ws)
    for (int k = 0; k < din; k += 4) {
        const float b0 = WcB[(long)k * dout];
        const float b1 = WcB[(long)k * dout + dout];
        v2f b;
        b.x = colok ? b0 : 0.0f;
        b.y = colok ? b1 : 0.0f;
        #pragma unroll
        for (int t = 0; t < MT; ++t) {
            float2 av = *(const float2*)(hrow[t] + k);
            v2f a;
            a.x = av.x;
            a.y = av.y;
            c[t] = __builtin_amdgcn_wmma_f32_16x16x4_f32(false, a, false, b, (short)0,
                                                         c[t], false, false);
        }
    }

    // epilogue: bias + ReLU + store, with fused BN column partial sums
    const float bias = colok ? (bl[colc] + bs[colc]) : 0.0f;
    float s = 0.0f, s2 = 0.0f;
    if (colok) {
        #pragma unroll
        for (int t = 0; t < MT; ++t) {
            #pragma unroll
            for (int r = 0; r < 8; ++r) {
                const int row = m0 + t * 16 + r + half * 8; // C/D: VGPR r -> M=r (+8 hi half)
                float v = c[t][r] + bias;
                if (do_relu) v = fmaxf(v, 0.0f);
                out[(long)row * dout + col] = v;
                s += v;
                s2 += v * v;
            }
        }
        if (do_bn) {
            atomicAdd(&colsum[col], s);
            atomicAdd(&colsq[col], s2);
        }
    }
}

// ---------------- BatchNorm apply (training-mode, biased var) ----------------

__global__ void bn_apply_kernel(float* __restrict__ h, const float* __restrict__ sum,
                                const float* __restrict__ sq, const float* __restrict__ g,
                                const float* __restrict__ be, int n, int dout) {
    long tid = (long)blockIdx.x * blockDim.x + threadIdx.x;
    long total = (long)n * dout;
    if (tid >= total) return;
    int col = (int)(tid % dout);
    float inv_n = 1.0f / (float)n;
    float mu = sum[col] * inv_n;
    float var = sq[col] * inv_n - mu * mu;
    float v = h[tid];
    h[tid] = (v - mu) * rsqrtf(var + BN_EPS) * g[col] + be[col];
}

// ---------------- host orchestration ----------------

extern "C" void kernel_launch(void* const* d_in, const int* in_sizes, int n_in,
                              void* d_out, int out_size, void* d_ws, size_t ws_size,
                              hipStream_t stream) {
    const float* x = (const float*)d_in[0];
    const int* ei = (const int*)d_in[1];
    const int E = in_sizes[1] / 2;
    const int* src = ei;       // edge_index[0]
    const int* dst = ei + E;   // edge_index[1]

    struct Layer { const float *Wl, *bl, *Wr, *Ws, *bs, *g, *be; int din, dout; };
    Layer ls[4];
    const int dims[5] = {128, 256, 128, 64, 40};
    int idx = 2;
    for (int li = 0; li < 4; ++li) {
        ls[li].Wl = (const float*)d_in[idx++];
        ls[li].bl = (const float*)d_in[idx++];
        ls[li].Wr = (const float*)d_in[idx++];
        ls[li].Ws = (const float*)d_in[idx++];
        ls[li].bs = (const float*)d_in[idx++];
        if (li < 3) {
            ls[li].g  = (const float*)d_in[idx++];
            ls[li].be = (const float*)d_in[idx++];
        } else {
            ls[li].g = nullptr;
            ls[li].be = nullptr;
        }
        ls[li].din = dims[li];
        ls[li].dout = dims[li + 1];
    }

    const int N = N_NODES;
    float* ws = (float*)d_ws;
    float* cnt    = ws;                             // N floats (padded)
    float* agg    = cnt + ((N + 255) / 256) * 256;  // N*256
    float* hA     = agg + (long)N * 256;            // N*256 (h1, later h3)
    float* hB     = hA + (long)N * 256;             // N*128 (h2)
    float* colsum = hB + (long)N * 128;             // 256
    float* colsq  = colsum + 256;                   // 256
    float* Wcbuf  = colsq + 256;                    // max din*dout = 32768

    // in-degree counts (once; shared by all layers)
    fill_kernel<<<(N + 255) / 256, 256, 0, stream>>>(cnt, 0.0f, N);
    degree_kernel<<<(E + 255) / 256, 256, 0, stream>>>(dst, cnt, E);

    const float* hin = x;
    float* houts[4] = {hA, hB, hA, (float*)d_out};

    for (int li = 0; li < 4; ++li) {
        const int din = ls[li].din, dout = ls[li].dout;
        const int do_bn = (li < 3) ? 1 : 0;

        // 1) zero + scatter-add neighbor sums into agg (L2-resident traffic)
        const int aggN = N * din;
        fill_kernel<<<(aggN + 255) / 256, 256, 0, stream>>>(agg, 0.0f, aggN);
        const long sthreads = (long)E * (din / 4);
        scatter_kernel<<<(unsigned)((sthreads + 255) / 256), 256, 0, stream>>>(
            hin, src, dst, agg, E, din);

        // 2) combined root+skip weight: Wc = Wr + Ws
        const int wn = din * dout;
        addvec_kernel<<<(wn + 255) / 256, 256, 0, stream>>>(ls[li].Wr, ls[li].Ws,
                                                            Wcbuf, wn);

        // 3) zero BN accumulators (GEMM epilogue fills them)
        if (do_bn) {
            fill_kernel<<<1, 256, 0, stream>>>(colsum, 0.0f, 256);
            fill_kernel<<<1, 256, 0, stream>>>(colsq, 0.0f, 256);
        }

        // 4) fused WMMA GEMM: (agg/cnt)@Wl + h@Wc + bias (+ReLU, +BN partials)
        dim3 grid(N / (16 * MT), (dout + 15) / 16);
        sage_gemm_kernel<<<grid, 32, 0, stream>>>(agg, hin, cnt, ls[li].Wl, Wcbuf,
                                                  ls[li].bl, ls[li].bs, houts[li],
                                                  colsum, colsq, din, dout,
                                                  do_bn, do_bn);

        // 5) BatchNorm normalize in-place
        if (do_bn) {
            const long tot = (long)N * dout;
            bn_apply_kernel<<<(unsigned)((tot + 255) / 256), 256, 0, stream>>>(
                houts[li], colsum, colsq, ls[li].g, ls[li].be, N, dout);
        }
        hin = houts[li];
    }
}